// PyramidBox_5669356832495
// MI455X (gfx1250) — compile-verified
//
#include <hip/hip_runtime.h>
#include <hip/hip_bf16.h>
#include <cstdint>

#define NUMC   21
#define TOPK   200
#define KP     208          // TOPK padded to 13*16
#define NPRI   120000
#define NB     8
#define CONF_T 0.05f
#define NMS_T  0.3f
#define NEG_INF (-__builtin_inff())

typedef float v2f __attribute__((ext_vector_type(2)));
typedef float v8f __attribute__((ext_vector_type(8)));

__device__ __forceinline__ unsigned mono_key(float v) {
  unsigned u = __float_as_uint(v);
  return u ^ ((u >> 31) ? 0xFFFFFFFFu : 0x80000000u);
}
__device__ __forceinline__ float unmono(unsigned k) {
  unsigned u = (k & 0x80000000u) ? (k ^ 0x80000000u) : ~k;
  return __uint_as_float(u);
}
__device__ __forceinline__ unsigned ballot32(bool p) {
#if __has_builtin(__builtin_amdgcn_ballot_w32)
  return __builtin_amdgcn_ballot_w32(p);
#else
  return (unsigned)__ballot(p);
#endif
}
__device__ __forceinline__ void wait_async0() {
#if __has_builtin(__builtin_amdgcn_s_wait_asynccnt)
  __builtin_amdgcn_s_wait_asynccnt(0);
#else
  asm volatile("s_wait_asynccnt 0" ::: "memory");
#endif
}
__device__ __forceinline__ void async_g2lds_b128(unsigned lds_off, const void* gaddr) {
  asm volatile("global_load_async_to_lds_b128 %0, %1, off"
               :: "v"(lds_off), "v"(gaddr) : "memory");
}

// ---------------------------------------------------------------------------
// Kernel T: transpose conf [B*P, 21] -> confT [B, 21, P], staged via async LDS
// 96 rows x 21 cols per block (96*21*4 = 8064 B = 504 x b128 chunks)
// ---------------------------------------------------------------------------
__global__ __launch_bounds__(256) void transpose_kernel(const float* __restrict__ conf,
                                                        float* __restrict__ confT) {
  __shared__ __align__(16) float tile[96 * NUMC];
  const int tid = threadIdx.x;
  const int blk = blockIdx.x;                    // 10000 blocks, 96 rows each
  const size_t base = (size_t)blk * 96 * NUMC;   // element offset into conf

#if defined(__gfx1250__)
  const unsigned lbase = (unsigned)(uintptr_t)(&tile[0]);
  for (int ch = tid; ch < 504; ch += 256) {
    async_g2lds_b128(lbase + (unsigned)ch * 16u, conf + base + (size_t)ch * 4);
  }
  wait_async0();
#else
  for (int e = tid; e < 96 * NUMC; e += 256) tile[e] = conf[base + e];
#endif
  __syncthreads();

  const int b  = blk / (NPRI / 96);
  const int p0 = (blk % (NPRI / 96)) * 96;
  for (int e = tid; e < 96 * NUMC; e += 256) {
    const int cls = e / 96;
    const int r   = e - cls * 96;
    confT[((size_t)b * NUMC + cls) * NPRI + p0 + r] = tile[r * NUMC + cls];
  }
}

// ---------------------------------------------------------------------------
// Kernel A: exact top-200 per (b,c) via 3-level radix select + bitonic sort
// grid (20, 8): c = blockIdx.x+1 (class 0 skipped), 256 threads
// ---------------------------------------------------------------------------
__global__ __launch_bounds__(256) void topk_kernel(const float* __restrict__ confT,
                                                   float* __restrict__ top_score,
                                                   int* __restrict__ top_idx) {
  const int c   = blockIdx.x + 1;
  const int b   = blockIdx.y;
  const int tid = threadIdx.x;
  const int lane = tid & 31;
  const float* cp = confT + ((size_t)b * NUMC + c) * NPRI;

  __shared__ unsigned hist[2048];
  __shared__ unsigned long long cands[512];
  __shared__ unsigned sh_prefix, sh_remaining, sh_cnt;

  if (tid == 0) { sh_prefix = 0u; sh_remaining = TOPK; }

  const int shifts[3]  = {21, 10, 0};
  const int widths[3]  = {11, 11, 10};

  for (int lvl = 0; lvl < 3; ++lvl) {
    const int shift = shifts[lvl];
    const int bins  = 1 << widths[lvl];
    const int chk   = (lvl == 0) ? 0 : shifts[lvl - 1];
    for (int e = tid; e < bins; e += 256) hist[e] = 0u;
    __syncthreads();
    const unsigned pref = sh_prefix;
    for (int p = tid; p < NPRI; p += 256) {
      const float v = cp[p];
      int bin = 0;
      bool active = false;
      if (v > CONF_T) {
        const unsigned key = mono_key(v);
        if (lvl == 0 || (key >> chk) == pref) {
          active = true;
          bin = (int)((key >> shift) & (unsigned)(bins - 1));
        }
      }
      // wave-aggregated histogram update (defuses same-bin LDS atomic storms)
      unsigned m = ballot32(active);
      while (m) {
        const int leader = __ffs(m) - 1;
        const int lbin = __shfl(bin, leader, 32);
        const unsigned smask = ballot32(active && (bin == lbin));
        if (lane == leader) atomicAdd(&hist[lbin], (unsigned)__popc(smask));
        m &= ~smask;
      }
    }
    __syncthreads();
    if (tid == 0) {
      const unsigned rem = sh_remaining;
      unsigned acc = 0, above = 0; int chosen = 0;
      for (int bin = bins - 1; bin >= 0; --bin) {
        const unsigned h = hist[bin];
        if (acc + h >= rem || bin == 0) { chosen = bin; above = acc; break; }
        acc += h;
      }
      sh_remaining = rem - above;
      sh_prefix = (sh_prefix << widths[lvl]) | (unsigned)chosen;
    }
    __syncthreads();
  }

  const unsigned thr = sh_prefix;   // exact key of the 200th largest candidate
  for (int e = tid; e < 512; e += 256) cands[e] = 0ull;
  if (tid == 0) sh_cnt = 0u;
  __syncthreads();
  for (int p = tid; p < NPRI; p += 256) {
    const float v = cp[p];
    if (v > CONF_T) {
      const unsigned key = mono_key(v);
      if (key >= thr) {
        const unsigned pos = atomicAdd(&sh_cnt, 1u);
        if (pos < 512u)
          cands[pos] = ((unsigned long long)key << 32) | (unsigned)(~(unsigned)p);
      }
    }
  }
  __syncthreads();

  // bitonic sort, descending: key desc, then idx asc (idx stored bit-inverted)
  for (int k = 2; k <= 512; k <<= 1) {
    for (int j = k >> 1; j > 0; j >>= 1) {
      for (int e = tid; e < 512; e += 256) {
        const int l = e ^ j;
        if (l > e) {
          const unsigned long long x = cands[e], y = cands[l];
          const bool desc = ((e & k) == 0);
          if (desc ? (x < y) : (x > y)) { cands[e] = y; cands[l] = x; }
        }
      }
      __syncthreads();
    }
  }

  if (tid < TOPK) {
    const unsigned long long cmb = cands[tid];
    const unsigned key = (unsigned)(cmb >> 32);
    float sc; int oi;
    if (key == 0u) { sc = NEG_INF; oi = 0; }
    else           { sc = unmono(key); oi = (int)(~(unsigned)cmb); }
    const size_t base = ((size_t)b * NUMC + c) * TOPK + tid;
    top_score[base] = sc;
    top_idx[base]   = oi;
  }
}

// ---------------------------------------------------------------------------
// Kernel B: decode 200 candidate boxes, build 208x208 suppression bitmask
// (WMMA outer-sum for area_i+area_j + VALU intersection), then single-wave NMS.
// grid (21, 8), 256 threads.
// ---------------------------------------------------------------------------
__global__ __launch_bounds__(256) void nms_kernel(const float* __restrict__ loc,
                                                  const float* __restrict__ prior,
                                                  const float* __restrict__ top_score,
                                                  const int* __restrict__ top_idx,
                                                  float* __restrict__ out) {
  const int c   = blockIdx.x;
  const int b   = blockIdx.y;
  const int tid = threadIdx.x;
  float* outp = out + ((size_t)b * NUMC + c) * TOPK * 5;

  if (c == 0) {   // reference overwrites background class with zeros
    for (int t = tid; t < TOPK * 5; t += 256) outp[t] = 0.f;
    return;
  }

  __shared__ float sx1[KP], sy1[KP], sx2[KP], sy2[KP], sarea[KP], sts[KP];
  __shared__ int   sidx[KP];
  __shared__ unsigned supmask[KP * 8];          // bit j of row i: iou(i,j) > 0.3
  __shared__ __align__(16) float stage[KP * 8]; // per-candidate {loc[4], prior[4]}

  if (tid < KP) {
    if (tid < TOPK) {
      const size_t base = ((size_t)b * NUMC + c) * TOPK + tid;
      sts[tid]  = top_score[base];
      sidx[tid] = top_idx[base];
    } else { sts[tid] = NEG_INF; sidx[tid] = 0; }
  }
  for (int e = tid; e < KP * 8; e += 256) supmask[e] = 0u;
  __syncthreads();

  // ---- gather loc/prior for my candidate into LDS (async per-lane gather) --
  const int t  = (tid < KP) ? tid : (KP - 1);
  const int gi = sidx[t];
  const float* lp = loc   + ((size_t)b * NPRI + (size_t)gi) * 4;
  const float* pp = prior + (size_t)gi * 4;
#if defined(__gfx1250__)
  {
    const unsigned l0 = (unsigned)(uintptr_t)(&stage[t * 8]);
    async_g2lds_b128(l0,      lp);
    async_g2lds_b128(l0 + 16, pp);
    wait_async0();
  }
  const float lx = stage[t*8+0], ly = stage[t*8+1], lw = stage[t*8+2], lh = stage[t*8+3];
  const float pcx = stage[t*8+4], pcy = stage[t*8+5], pw = stage[t*8+6], ph = stage[t*8+7];
#else
  const float lx = lp[0], ly = lp[1], lw = lp[2], lh = lp[3];
  const float pcx = pp[0], pcy = pp[1], pw = pp[2], ph = pp[3];
#endif
  {
    const float cx = pcx + lx * 0.1f * pw;
    const float cy = pcy + ly * 0.1f * ph;
    const float w  = pw * expf(lw * 0.2f);
    const float h  = ph * expf(lh * 0.2f);
    const float x1 = cx - w * 0.5f, y1 = cy - h * 0.5f;
    const float x2 = x1 + w,        y2 = y1 + h;
    if (tid < KP) {
      sx1[t] = x1; sy1[t] = y1; sx2[t] = x2; sy2[t] = y2;
      sarea[t] = (x2 - x1) * (y2 - y1);
    }
  }
  __syncthreads();

  // ---- 13x13 tiles of 16x16: area_i+area_j via WMMA, intersection via VALU -
  const int lane = tid & 31;
  const int wv   = tid >> 5;
  const int halfAdd = (lane < 16) ? 0 : 8;
  for (int tile = wv; tile < 169; tile += 8) {
    const int ti = tile / 13, tj = tile - ti * 13;
    const int rbase = ti * 16, cbase = tj * 16;
    v2f A, Bm;
    A.x  = (lane < 16) ? sarea[rbase + lane] : 0.f;  // A K=0 row: area_i
    A.y  = (lane < 16) ? 1.0f : 0.f;                 // A K=1 row: ones
    Bm.x = (lane < 16) ? 1.0f : 0.f;                 // B K=0 row: ones
    Bm.y = (lane < 16) ? sarea[cbase + lane] : 0.f;  // B K=1 row: area_j
    v8f Cz = {0.f,0.f,0.f,0.f,0.f,0.f,0.f,0.f};
    v8f D = __builtin_amdgcn_wmma_f32_16x16x4_f32(false, A, false, Bm,
                                                  (short)0, Cz, false, false);
    const int col = cbase + (lane & 15);
    const float cx1 = sx1[col], cy1 = sy1[col], cx2 = sx2[col], cy2 = sy2[col];
#pragma unroll
    for (int r = 0; r < 8; ++r) {
      const int row = rbase + r + halfAdd;
      const float xx1 = fmaxf(sx1[row], cx1);
      const float yy1 = fmaxf(sy1[row], cy1);
      const float xx2 = fminf(sx2[row], cx2);
      const float yy2 = fminf(sy2[row], cy2);
      const float w = fmaxf(xx2 - xx1, 0.f);
      const float h = fmaxf(yy2 - yy1, 0.f);
      const float inter = w * h;
      const float uni   = D[r] - inter;        // (area_i+area_j) - inter
      // iou > T  <=>  inter > T*uni  (uni >= 0 geometrically; avoids the
      // IEEE v_div_scale/v_rcp/v_div_fmas/v_div_fixup sequence per element)
      const bool  sup   = inter > NMS_T * uni;
      const unsigned bal = ballot32(sup);      // bits 0-15: row r; 16-31: row r+8
      if (lane == 0) {
        const int word = tj >> 1;
        const unsigned sh = (unsigned)(tj & 1) * 16u;
        atomicOr(&supmask[(rbase + r)     * 8 + word], (bal & 0xFFFFu) << sh);
        atomicOr(&supmask[(rbase + r + 8) * 8 + word], (bal >> 16)     << sh);
      }
    }
  }
  __syncthreads();

  // ---- sequential NMS in a single wave: no block barriers in the hot loop --
  if (wv == 0) {
    float rs[7];
#pragma unroll
    for (int q = 0; q < 7; ++q) {
      const int j = lane + (q << 5);
      rs[q] = (j < KP) ? sts[j] : NEG_INF;
    }
    for (int step = 0; step < TOPK; ++step) {
      float bv = rs[0]; int bj = lane;
#pragma unroll
      for (int q = 1; q < 7; ++q) {
        const int j = lane + (q << 5);
        if (rs[q] > bv) { bv = rs[q]; bj = j; }
      }
      for (int off = 16; off > 0; off >>= 1) {
        const float ov = __shfl_down(bv, off, 32);
        const int   oj = __shfl_down(bj, off, 32);
        if (ov > bv || (ov == bv && oj < bj)) { bv = ov; bj = oj; }
      }
      const int   i    = __shfl(bj, 0, 32);
      const float bval = __shfl(bv, 0, 32);
      const bool  ok   = bval > NEG_INF;
      if (lane < 5) {
        const float val = (lane == 0) ? sts[i] : (lane == 1) ? sx1[i]
                        : (lane == 2) ? sy1[i] : (lane == 3) ? sx2[i] : sy2[i];
        outp[step * 5 + lane] = ok ? val : 0.f;
      }
#pragma unroll
      for (int q = 0; q < 7; ++q) {
        const int j = lane + (q << 5);
        const unsigned wbits = supmask[i * 8 + q];
        if ((((wbits >> lane) & 1u) != 0u) || (j == i)) rs[q] = NEG_INF;
      }
    }
  }
}

// ---------------------------------------------------------------------------
extern "C" void kernel_launch(void* const* d_in, const int* in_sizes, int n_in,
                              void* d_out, int out_size, void* d_ws, size_t ws_size,
                              hipStream_t stream) {
  (void)in_sizes; (void)n_in; (void)out_size; (void)ws_size;
  const float* loc   = (const float*)d_in[0];
  const float* conf  = (const float*)d_in[1];
  const float* prior = (const float*)d_in[2];
  float* out = (float*)d_out;

  // workspace layout: confT (B*21*P f32) | top_score (B*21*200 f32) | top_idx
  float* confT     = (float*)d_ws;
  float* top_score = confT + (size_t)NB * NUMC * NPRI;
  int*   top_idx   = (int*)(top_score + (size_t)NB * NUMC * TOPK);

  transpose_kernel<<<dim3((NB * NPRI) / 96), 256, 0, stream>>>(conf, confT);
  topk_kernel<<<dim3(NUMC - 1, NB), 256, 0, stream>>>(confT, top_score, top_idx);
  nms_kernel<<<dim3(NUMC, NB), 256, 0, stream>>>(loc, prior, top_score, top_idx, out);
}